// GPUDataAugmentation_67645734912530
// MI455X (gfx1250) — compile-verified
//
#include <hip/hip_runtime.h>

// ---------------- problem constants (match reference) ----------------
#define BN   256
#define CN   3
#define HN   224
#define WN   224
#define NPER   (CN * HN * WN)     // 150528 elements per sample
#define NPER4  (NPER / 4)         // 37632 float4 per sample
#define HW4    (HN * WN / 4)      // 12544 float4 per channel plane
#define W4     (WN / 4)           // 56 float4 per row
#define TPB    256
#define SPLIT  21                 // 37632 = 21 * 256 * 7
#define VPT    (NPER4 / (SPLIT * TPB))   // 7 float4 per thread in pass 1
#define BLKS_PER_SAMPLE (NPER4 / TPB)    // 147

typedef float f4 __attribute__((ext_vector_type(4)));

struct __align__(16) SampleParams {
    float bscale;   // brightness multiplier (1.0 if not applied)
    float cscale;   // contrast multiplier
    float mean;     // post-brightness per-sample mean
    int   flags;    // bit0 = flip, bit1 = contrast-applied
    int   ph, pw, eh, ew;  // erase box (eh=ew=0 if no erase)
};

// ---------------- pass 1: per-sample partial sums (deterministic) ----
__global__ void aug_sum_kernel(const float* __restrict__ x,
                               float* __restrict__ partials) {
    const int b = blockIdx.x;       // sample
    const int s = blockIdx.y;       // split
    const int t = threadIdx.x;

    const f4* __restrict__ xb = (const f4*)(x + (size_t)b * NPER);

    float acc = 0.0f;
    #pragma unroll
    for (int k = 0; k < VPT; ++k) {
        // strided so consecutive threads touch consecutive float4 (coalesced b128)
        int i4 = k * (SPLIT * TPB) + s * TPB + t;
        f4 v = xb[i4];
        acc += (v.x + v.y) + (v.z + v.w);
    }

    __shared__ float red[TPB];
    red[t] = acc;
    __syncthreads();
    #pragma unroll
    for (int off = TPB / 2; off > 0; off >>= 1) {
        if (t < off) red[t] += red[t + off];
        __syncthreads();
    }
    if (t == 0) partials[b * SPLIT + s] = red[0];
}

// ---------------- pass 2a: fold RNG draws into per-sample params -----
__global__ void aug_param_kernel(const float* __restrict__ partials,
                                 const float* __restrict__ flip_u,
                                 const float* __restrict__ bright_u,
                                 const float* __restrict__ bright_v,
                                 const float* __restrict__ contrast_u,
                                 const float* __restrict__ contrast_v,
                                 const float* __restrict__ erase_u,
                                 const float* __restrict__ erase_box_u,
                                 SampleParams* __restrict__ params) {
    const int b = threadIdx.x;     // one block of 256 threads
    if (b >= BN) return;

    // deterministic fixed-order sum of the 21 partials
    float sum = 0.0f;
    #pragma unroll
    for (int s = 0; s < SPLIT; ++s) sum += partials[b * SPLIT + s];
    float mean_raw = sum * (1.0f / (float)NPER);

    const bool flip = flip_u[b]     < 0.5f;
    const bool bm   = bright_u[b]   < 0.7f;
    const bool cm   = contrast_u[b] < 0.7f;
    const bool em   = erase_u[b]    < 0.3f;

    const float bscale = bm ? (0.8f + 0.4f * bright_v[b])   : 1.0f;
    const float cscale = cm ? (0.8f + 0.4f * contrast_v[b]) : 1.0f;
    // flip preserves the mean; brightness scales it uniformly
    const float mean = bm ? mean_raw * bscale : mean_raw;

    int eh = (int)floorf((float)HN * (0.02f + 0.18f * erase_box_u[b * 4 + 0]));
    int ew = (int)floorf((float)WN * (0.02f + 0.18f * erase_box_u[b * 4 + 1]));
    int ph = (int)floorf((float)(HN - eh) * erase_box_u[b * 4 + 2]);
    int pw = (int)floorf((float)(WN - ew) * erase_box_u[b * 4 + 3]);
    if (!em) { eh = 0; ew = 0; ph = 0; pw = 0; }   // empty box

    SampleParams p;
    p.bscale = bscale;
    p.cscale = cscale;
    p.mean   = mean;
    p.flags  = (flip ? 1 : 0) | (cm ? 2 : 0);
    p.ph = ph; p.pw = pw; p.eh = eh; p.ew = ew;
    params[b] = p;
}

// ---------------- pass 2b: elementwise apply (b128 in, NT b128 out) --
__global__ void aug_apply_kernel(const float* __restrict__ x,
                                 const SampleParams* __restrict__ params,
                                 float* __restrict__ out) {
    const int b  = blockIdx.y;                       // sample (block-uniform)
    const int i4 = blockIdx.x * TPB + threadIdx.x;   // float4 index within sample

    const SampleParams p = params[b];                // uniform -> scalar loads
    const bool flip = (p.flags & 1) != 0;
    const bool cm   = (p.flags & 2) != 0;

    // decode (channel is irrelevant: all transforms uniform across C)
    const int hw4 = i4 % HW4;
    const int h   = hw4 / W4;
    const int w4  = hw4 % W4;
    const int w   = w4 * 4;

    const f4* __restrict__ xb = (const f4*)(x + (size_t)b * NPER);

    // horizontal flip: mirrored float4 + lane reversal (W % 4 == 0 keeps alignment)
    int src4 = flip ? (i4 - w4 + (W4 - 1 - w4)) : i4;
    f4 v = xb[src4];
    if (flip) { f4 r = { v.w, v.z, v.y, v.x }; v = r; }

    // brightness (bscale==1.0f is an exact identity)
    v *= p.bscale;

    // contrast around post-brightness mean
    if (cm) {
        v = (v - p.mean) * p.cscale + p.mean;
    }

    // random erase on output coordinates
    const bool rowin = (h >= p.ph) & (h < p.ph + p.eh);
    if (rowin) {
        #pragma unroll
        for (int jj = 0; jj < 4; ++jj) {
            int col = w + jj;
            if (col >= p.pw && col < p.pw + p.ew) v[jj] = 0.0f;
        }
    }

    // clip to [0,1]
    #pragma unroll
    for (int jj = 0; jj < 4; ++jj)
        v[jj] = fminf(fmaxf(v[jj], 0.0f), 1.0f);

    // non-temporal store: keep x resident in the 192MB L2, stream output to HBM
    f4* __restrict__ ob = (f4*)(out + (size_t)b * NPER);
    __builtin_nontemporal_store(v, &ob[i4]);
}

// ---------------- host-side launch ----------------------------------
extern "C" void kernel_launch(void* const* d_in, const int* in_sizes, int n_in,
                              void* d_out, int out_size, void* d_ws, size_t ws_size,
                              hipStream_t stream) {
    const float* x          = (const float*)d_in[0];
    const float* flip_u     = (const float*)d_in[1];
    const float* bright_u   = (const float*)d_in[2];
    const float* bright_v   = (const float*)d_in[3];
    const float* contrast_u = (const float*)d_in[4];
    const float* contrast_v = (const float*)d_in[5];
    const float* erase_u    = (const float*)d_in[6];
    const float* erase_box  = (const float*)d_in[7];
    float* out = (float*)d_out;

    // workspace layout: [0, 21KB) partial sums; [32KB, 40KB) params
    float*         partials = (float*)d_ws;
    SampleParams*  params   = (SampleParams*)((char*)d_ws + (32 << 10));

    // pass 1: deterministic per-sample sums (fills L2 with x)
    aug_sum_kernel<<<dim3(BN, SPLIT), TPB, 0, stream>>>(x, partials);

    // pass 2a: per-sample parameter folding
    aug_param_kernel<<<1, TPB, 0, stream>>>(partials, flip_u, bright_u, bright_v,
                                            contrast_u, contrast_v, erase_u,
                                            erase_box, params);

    // pass 2b: elementwise apply, one block per (sample, 1024-elem chunk)
    aug_apply_kernel<<<dim3(BLKS_PER_SAMPLE, BN), TPB, 0, stream>>>(x, params, out);
}